// PhaseSynchronizationLoss_71683004170569
// MI455X (gfx1250) — compile-verified
//
#include <hip/hip_runtime.h>
#include <math.h>

constexpr int   B_DIM = 8;
constexpr int   S_LEN = 4096;
constexpr int   D_DIM = 512;
constexpr int   CPG   = 4;                 // channels per workgroup (b128 = 4 floats along D)
constexpr int   TPB   = 256;               // 8 wave32
constexpr int   EPT   = S_LEN / TPB;       // 16 elements per thread
constexpr int   NWG   = B_DIM * (D_DIM / CPG);  // 1024 workgroups
constexpr float PI_F  = 3.14159265358979323846f;
constexpr float TPI_F = 6.28318530717958647692f;

// Per-thread chunk = EPT rows * CPG words = 64 words; pad +4 words (16B) per chunk so the
// wave-wide word stride (68) is not a multiple of the 64 LDS banks: 32-way -> 2-way conflicts.
// Rows stay 16B-aligned & contiguous, as required by the async b128 LDS destination.
constexpr int CHUNK_W = EPT * CPG + 4;             // 68 words per 16-row chunk
constexpr int TILE_W  = (S_LEN / EPT) * CHUNK_W;   // 17408 words = 69632 B per tile

__device__ __forceinline__ int sidx(int s, int c) {   // word index of element (s, c) in a tile
  return (s >> 4) * CHUNK_W + (s & 15) * CPG + c;
}

__device__ __forceinline__ unsigned lds_off_of(const void* p) {
  // low 32 bits of a flat LDS pointer = group-segment byte offset (ISA 10.2 aperture rule)
  return (unsigned)(size_t)p;
}

__device__ __forceinline__ void async_load_b128(unsigned lds_off, const float* g) {
  // CDNA5 async copy global->LDS, tracked on ASYNCcnt (one 16B element per lane)
  asm volatile("global_load_async_to_lds_b128 %0, %1, off"
               :: "v"(lds_off), "v"(g) : "memory");
}

__device__ __forceinline__ float block_sum_f32(float v, float* s_red, int tid) {
  #pragma unroll
  for (int off = 16; off > 0; off >>= 1) v += __shfl_xor(v, off, 32);   // wave32 tree
  const int w = tid >> 5;
  if ((tid & 31) == 0) s_red[w] = v;
  __syncthreads();
  float tot = 0.f;
  #pragma unroll
  for (int i = 0; i < TPB / 32; ++i) tot += s_red[i];   // every thread: deterministic broadcast
  __syncthreads();
  return tot;
}

__device__ __forceinline__ int block_exscan_i32(int v, int* s_red, int tid, int* total) {
  int incl = v;
  #pragma unroll
  for (int off = 1; off < 32; off <<= 1) {               // wave32 inclusive scan
    int u = __shfl_up(incl, off, 32);
    if ((tid & 31) >= off) incl += u;
  }
  const int w = tid >> 5;
  if ((tid & 31) == 31) s_red[w] = incl;
  __syncthreads();
  int woff = 0, tot = 0;
  #pragma unroll
  for (int i = 0; i < TPB / 32; ++i) {
    int x = s_red[i];
    if (i < w) woff += x;
    tot += x;
  }
  __syncthreads();
  *total = tot;
  return woff + incl - v;
}

// Reference-faithful zero-crossing phase for channel c of a padded-interleaved LDS tile.
// Leaves the EPT phases of this thread's contiguous element chunk in ph[].
__device__ void channel_phases(const float* __restrict__ sbuf, int c, int tid,
                               int* __restrict__ s_zc, float* s_redf, int* s_redi,
                               float ph[EPT]) {
  const int base = tid * EPT;
  float x[EPT];
  #pragma unroll
  for (int j = 0; j < EPT; ++j) x[j] = sbuf[sidx(base + j, c)];
  const float xn = (base + EPT < S_LEN) ? sbuf[sidx(base + EPT, c)] : 0.f;

  float lsum = 0.f;
  #pragma unroll
  for (int j = 0; j < EPT; ++j) lsum += x[j];
  const float mean = block_sum_f32(lsum, s_redf, tid) * (1.f / (float)S_LEN);

  bool sb[EPT + 1];
  #pragma unroll
  for (int j = 0; j < EPT; ++j) sb[j] = __builtin_signbitf(x[j] - mean);
  sb[EPT] = __builtin_signbitf(xn - mean);

  // crossings at positions i in [base, base+EPT) with i <= S-2
  int lc = 0;
  #pragma unroll
  for (int j = 0; j < EPT; ++j)
    if (base + j < S_LEN - 1 && sb[j] != sb[j + 1]) ++lc;

  int n;
  const int pfx = block_exscan_i32(lc, s_redi, tid, &n);

  // walk: record per-element crossing count (== reference ccount[t]) and compact positions
  int run = pfx;
  int cnt[EPT];
  #pragma unroll
  for (int j = 0; j < EPT; ++j) {
    cnt[j] = run;  // #crossings at positions < t
    if (base + j < S_LEN - 1 && sb[j] != sb[j + 1]) { s_zc[run] = base + j; ++run; }
  }
  __syncthreads();

  if (n < 2) {   // uniform across block
    #pragma unroll
    for (int j = 0; j < EPT; ++j) ph[j] = 0.f;
    return;
  }

  const int   zfirst     = s_zc[0];
  const int   zlast      = s_zc[n - 1];
  const float nf         = (float)n;
  const float avg_period = ((float)zlast - (float)zfirst) / fmaxf(nf - 1.f, 1.f);
  const int   last_final = zlast + 1;
  const float remaining  = (float)(S_LEN - last_final);
  const float tail_slope = PI_F * remaining / fmaxf(avg_period, 1e-9f);
  const float tail_div   = fmaxf(remaining - 1.f, 1.f);

  #pragma unroll
  for (int j = 0; j < EPT; ++j) {
    const int t  = base + j;
    const int ct = cnt[j];
    float p;
    if (ct == n) {                       // tail <=> next_zc >= S
      const float ii = (float)(t - last_final);
      p = PI_F * nf + ((remaining > 1.f) ? tail_slope * ii / tail_div : 0.f);
    } else {                             // segment [prev_zc+1, next_zc]
      const int   prev  = (ct > 0) ? s_zc[ct - 1] : -1;
      const int   next  = s_zc[ct];
      const int   lastp = prev + 1;
      const int   L     = next - lastp + 1;
      const float frac  = (L > 1) ? (float)(t - lastp) / (float)(L - 1) : 0.f;
      p = PI_F * (float)ct + PI_F * frac;
    }
    // (p + pi) mod 2pi - pi, python floor-mod semantics
    float w = p + PI_F;
    w -= floorf(w / TPI_F) * TPI_F;
    ph[j] = w - PI_F;
  }
}

__global__ __launch_bounds__(TPB) void phase_sync_partials(
    const float* __restrict__ in_sig, const float* __restrict__ at_sig,
    float* __restrict__ part_c, float* __restrict__ part_s) {
  __shared__ __align__(16) float s_in[TILE_W];   // ~68 KB
  __shared__ __align__(16) float s_at[TILE_W];   // ~68 KB (raw, then phases in place)
  __shared__ int   s_zc[S_LEN];                  // 16 KB crossing positions
  __shared__ float s_redf[TPB / 32];
  __shared__ int   s_redi[TPB / 32];

  const int tid  = threadIdx.x;
  const int wg   = blockIdx.x;
  const int b    = wg / (D_DIM / CPG);
  const int d0   = (wg % (D_DIM / CPG)) * CPG;
  const size_t cbase = (size_t)b * S_LEN * D_DIM + (size_t)d0;
  const int base = tid * EPT;

  // stage both 4-channel tiles via async DMA to LDS; attractor first (used in pass 1)
  const unsigned l_at = lds_off_of(s_at);
  const unsigned l_in = lds_off_of(s_in);
  #pragma unroll
  for (int j = 0; j < EPT; ++j) {
    const int s = base + j;
    async_load_b128(l_at + (unsigned)(sidx(s, 0) * 4), at_sig + cbase + (size_t)s * D_DIM);
  }
  #pragma unroll
  for (int j = 0; j < EPT; ++j) {
    const int s = base + j;
    async_load_b128(l_in + (unsigned)(sidx(s, 0) * 4), in_sig + cbase + (size_t)s * D_DIM);
  }

  asm volatile("s_wait_asynccnt 16" ::: "memory");  // first 16 asyncs (attractor) complete
  __syncthreads();

  // pass 1: attractor phases, written back in place (raw values fully consumed first)
  for (int c = 0; c < CPG; ++c) {
    float ph[EPT];
    channel_phases(s_at, c, tid, s_zc, s_redf, s_redi, ph);
    #pragma unroll
    for (int j = 0; j < EPT; ++j) s_at[sidx(base + j, c)] = ph[j];
    __syncthreads();
  }

  asm volatile("s_wait_asynccnt 0" ::: "memory");   // input tile resident
  __syncthreads();

  // pass 2: input phases + accumulate cos/sin of phase difference
  float acc_c = 0.f, acc_s = 0.f;
  for (int c = 0; c < CPG; ++c) {
    float ph[EPT];
    channel_phases(s_in, c, tid, s_zc, s_redf, s_redi, ph);
    #pragma unroll
    for (int j = 0; j < EPT; ++j) {
      const float pd = ph[j] - s_at[sidx(base + j, c)];
      float sv, cv;
      __sincosf(pd, &sv, &cv);   // pd in (-2pi, 2pi): hw v_sin/v_cos path is accurate here
      acc_c += cv;
      acc_s += sv;
    }
    __syncthreads();
  }

  const float wc = block_sum_f32(acc_c, s_redf, tid);
  const float ws = block_sum_f32(acc_s, s_redf, tid);
  if (tid == 0) { part_c[wg] = wc; part_s[wg] = ws; }
}

__global__ __launch_bounds__(TPB) void phase_sync_finalize(
    const float* __restrict__ part_c, const float* __restrict__ part_s,
    float* __restrict__ out) {
  __shared__ float s_redf[TPB / 32];
  float c = 0.f, s = 0.f;
  for (int i = threadIdx.x; i < NWG; i += TPB) { c += part_c[i]; s += part_s[i]; }
  c = block_sum_f32(c, s_redf, threadIdx.x);
  s = block_sum_f32(s, s_redf, threadIdx.x);
  if (threadIdx.x == 0) {
    const float invN = 1.f / (float)((long long)B_DIM * S_LEN * D_DIM);
    const float mr = c * invN, mi = s * invN;
    out[0] = 0.1f * (1.f - sqrtf(mr * mr + mi * mi));
  }
}

extern "C" void kernel_launch(void* const* d_in, const int* in_sizes, int n_in,
                              void* d_out, int out_size, void* d_ws, size_t ws_size,
                              hipStream_t stream) {
  (void)in_sizes; (void)n_in; (void)out_size; (void)ws_size;
  const float* in_sig = (const float*)d_in[0];
  const float* at_sig = (const float*)d_in[1];
  float* part_c = (float*)d_ws;          // NWG floats
  float* part_s = part_c + NWG;          // NWG floats (8 KB total scratch)
  phase_sync_partials<<<NWG, TPB, 0, stream>>>(in_sig, at_sig, part_c, part_s);
  phase_sync_finalize<<<1, TPB, 0, stream>>>(part_c, part_s, (float*)d_out);
}